// DenseMap_39573828665602
// MI455X (gfx1250) — compile-verified
//
#include <hip/hip_runtime.h>
#include <stdint.h>

typedef __attribute__((ext_vector_type(2))) float f2;
typedef __attribute__((ext_vector_type(4))) float f4;

#define RES      128
#define MAPS     128
#define SAMPLES  32768
#define PAIRS    (SAMPLES * MAPS)   // 4,194,304
#define BLOCK    256                // 8 wave32 waves
#define ITER     4                  // pairs per thread
#define CHUNK    (BLOCK * ITER)     // 1024 pairs per block

__global__ __launch_bounds__(BLOCK)
void densemap_bilerp_kernel(const float* __restrict__ inputs,
                            const float* __restrict__ emb,
                            float* __restrict__ out)
{
    __shared__ f2    s_in[CHUNK];          // 8 KB staged coords
    __shared__ float s_out[CHUNK * 10];    // 40 KB staged output tile

    const int tid       = threadIdx.x;
    const int base_pair = (int)blockIdx.x * CHUNK;          // <= 4.19M: int32 ok
    const f2* __restrict__ gin = reinterpret_cast<const f2*>(inputs) + base_pair;

    // ---- Stage this block's input coords into LDS via CDNA5 async copies
    // (streamed once -> non-temporal load hint). Generic LDS pointers carry
    // the DS byte offset in their low 32 bits.
#pragma unroll
    for (int j = 0; j < ITER; ++j) {
        const int idx = j * BLOCK + tid;
        uint32_t  lds_addr = (uint32_t)(uintptr_t)&s_in[idx];
        const f2* gaddr    = gin + idx;
        asm volatile("global_load_async_to_lds_b64 %0, %1, off th:TH_LOAD_NT"
                     :
                     : "v"(lds_addr), "v"(gaddr)
                     : "memory");
    }
    asm volatile("s_wait_asynccnt 0x0" ::: "memory");
    __syncthreads();

    // ---- Phase 1: corner rows + fracs; prefetch all corner cachelines
    // (global_prefetch_b8) so the b128 gathers below have L2 lead time.
    int   r4a[ITER];
    float f0a[ITER], f1a[ITER];
#pragma unroll
    for (int j = 0; j < ITER; ++j) {
        const int idx = j * BLOCK + tid;
        f2    c  = s_in[idx];
        float x0 = c.x * (float)(RES - 1);
        float x1 = c.y * (float)(RES - 1);
        int   i0 = (int)x0;                  // inputs >= 0 -> trunc == floor
        int   i1 = (int)x1;
        f0a[j] = x0 - (float)i0;
        f1a[j] = x1 - (float)i1;
        // base_pair is a multiple of 128, so map == idx & 127 (pure 32-bit).
        int map = idx & (MAPS - 1);
        int row = map * (RES * RES) + i0 * RES + i1;
        r4a[j]  = row * 2;                   // offset in float4 units (32 B rows)
        const f4* e = reinterpret_cast<const f4*>(emb) + r4a[j];
        __builtin_prefetch(e,           0, 3);  // rows (i0, i1) & (i0, i1+1): 64 B
        __builtin_prefetch(e + 2 * RES, 0, 3);  // rows (i0+1, i1) & (i0+1, i1+1)
    }

    // ---- Phase 2: gather 4 corner rows (2x b128 each), bilerp, write tile
    // to LDS (conflict-free: 10-dword lane stride -> 32 distinct bank pairs).
#pragma unroll
    for (int j = 0; j < ITER; ++j) {
        const int idx = j * BLOCK + tid;
        const f4* e = reinterpret_cast<const f4*>(emb) + r4a[j];
        f4 a0 = e[0],           a1 = e[1];            // (i0  , i1  )
        f4 b0 = e[2],           b1 = e[3];            // (i0  , i1+1)
        f4 c0 = e[2 * RES],     c1 = e[2 * RES + 1];  // (i0+1, i1  )
        f4 d0 = e[2 * RES + 2], d1 = e[2 * RES + 3];  // (i0+1, i1+1)

        float f0 = f0a[j], f1 = f1a[j];
        float w00 = (1.0f - f0) * (1.0f - f1);
        float w01 = (1.0f - f0) * f1;
        float w10 = f0 * (1.0f - f1);
        float w11 = f0 * f1;

        f4 lo = a0 * w00 + b0 * w01 + c0 * w10 + d0 * w11;
        f4 hi = a1 * w00 + b1 * w01 + c1 * w10 + d1 * w11;

        f2* o = reinterpret_cast<f2*>(s_out + idx * 10);  // 8 B aligned
        o[0] = (f2){lo.x, lo.y};
        o[1] = (f2){lo.z, lo.w};
        o[2] = (f2){hi.x, hi.y};
        o[3] = (f2){hi.z, hi.w};
        o[4] = (f2){f0, f1};
    }
    __syncthreads();

    // ---- Phase 3: stream the 40 KB tile out as contiguous, fully coalesced
    // async b128 stores straight from LDS (CDNA5 data mover, no VGPR round
    // trip). NT hint keeps the 168 MB output stream from evicting the
    // L2-resident 64 MB embedding table.
    {
        float*   gout     = out + (int64_t)blockIdx.x * (CHUNK * 10); // 16B-aligned
        uint32_t lds_base = (uint32_t)(uintptr_t)s_out;
#pragma unroll
        for (int k = 0; k < (CHUNK * 10) / 4 / BLOCK; ++k) {   // 10 chunks/thread
            int       off4     = k * BLOCK + tid;              // float4 index
            uint32_t  lds_addr = lds_base + (uint32_t)off4 * 16u;
            const f4* gaddr    = reinterpret_cast<const f4*>(gout) + off4;
            asm volatile("global_store_async_from_lds_b128 %0, %1, off th:TH_STORE_NT"
                         :
                         : "v"(gaddr), "v"(lds_addr)
                         : "memory");
        }
    }
    asm volatile("s_wait_asynccnt 0x0" ::: "memory");
}

extern "C" void kernel_launch(void* const* d_in, const int* in_sizes, int n_in,
                              void* d_out, int out_size, void* d_ws, size_t ws_size,
                              hipStream_t stream) {
    const float* inputs = (const float*)d_in[0];  // (32768, 128, 2) f32
    const float* emb    = (const float*)d_in[1];  // (128*128*128, 8) f32
    float*       out    = (float*)d_out;          // (32768, 128, 10) f32

    const int blocks = PAIRS / CHUNK;             // 4096, exact (no tail)
    densemap_bilerp_kernel<<<dim3(blocks), dim3(BLOCK), 0, stream>>>(inputs, emb, out);
}